// BoundaryPointVotingModule_61856118997114
// MI455X (gfx1250) — compile-verified
//
#include <hip/hip_runtime.h>
#include <math.h>

#define NPTS   65536
#define MB     2048
#define KPARTS 8
#define KNN_N  16
#define RADIUS 0.03f
#define INV_RADIUS (1.0f / 0.03f)
#define WAVES  8                       // waves per block (256 threads)
#define CSTRIDE (WAVES * 16)           // column stride per block iteration
#define NCAND  (WAVES * 32)            // knn candidates per row (2 halves x 16 each)

typedef __attribute__((ext_vector_type(2))) float v2f;
typedef __attribute__((ext_vector_type(8))) float v8f;

__device__ __forceinline__ v8f wmma_f32_16x16x4(v2f a, v2f b, v8f c) {
  // D(16x16,f32) = A(16x4,f32) x B(4x16,f32) + C
  return __builtin_amdgcn_wmma_f32_16x16x4_f32(
      /*neg_a=*/false, a, /*neg_b=*/false, b,
      /*c_mod=*/(short)0, c, /*reuse_a=*/false, /*reuse_b=*/false);
}

// raw v_sqrt_f32: args are clamped >= 1e-12 (no denorms), ~1 ulp is plenty
__device__ __forceinline__ float fast_sqrt(float x) {
  return __builtin_amdgcn_sqrtf(x);
}

// ---------------------------------------------------------------------------
// Phase 1a: per-point argmax label over the 8 part logits
// ---------------------------------------------------------------------------
__global__ void labels_kernel(const float* __restrict__ part_mask,
                              int* __restrict__ labels) {
  int i = blockIdx.x * blockDim.x + threadIdx.x;
  if (i >= NPTS) return;
  const float* p = part_mask + i * KPARTS;
  int best = 0;
  float bv = p[0];
#pragma unroll
  for (int k = 1; k < KPARTS; ++k) {
    float v = p[k];
    if (v > bv) { bv = v; best = k; }   // first-max semantics like jnp.argmax
  }
  labels[i] = best;
}

// ---------------------------------------------------------------------------
// Phase 1b: kNN label-disagreement ratio -> boundary mask (written as float)
// mean >= 0.3 over 16 neighbors  <=>  mismatch count >= 5
// ---------------------------------------------------------------------------
__global__ void boundary_kernel(const int* __restrict__ labels,
                                const int* __restrict__ nbr,
                                float* __restrict__ out) {
  int i = blockIdx.x * blockDim.x + threadIdx.x;
  if (i >= NPTS) return;
  int l = labels[i];
  int cnt = 0;
#pragma unroll
  for (int j = 0; j < KNN_N; ++j)
    cnt += (labels[nbr[i * KNN_N + j]] != l) ? 1 : 0;
  out[i] = (cnt >= 5) ? 1.0f : 0.0f;
}

// ---------------------------------------------------------------------------
// Phase 1c: boundary_mask[boundary_indices] = cluster_labels + 1
// ---------------------------------------------------------------------------
__global__ void scatter_kernel(const int* __restrict__ bidx,
                               const int* __restrict__ clab,
                               float* __restrict__ out) {
  int m = blockIdx.x * blockDim.x + threadIdx.x;
  if (m >= MB) return;
  out[bidx[m]] = (float)(clab[m] + 1);
}

// ---------------------------------------------------------------------------
// Phase 2: local voting. One block = 16 boundary rows; 8 waves split columns.
// WMMA #1 computes the 16x16 d^2 tile (Gram trick), WMMA #2..#5 accumulate
// w @ [part_mask | 1] so column 8 of the accumulator is sum(w).
// ---------------------------------------------------------------------------
__global__ void __launch_bounds__(WAVES * 32)
voting_kernel(const float* __restrict__ points,
              const float* __restrict__ part_mask,
              const int* __restrict__ boundary_indices,
              const int* __restrict__ cluster_labels,
              float* __restrict__ out) {
  __shared__ float s_bx[16], s_by[16], s_bz[16], s_bsq[16];
  __shared__ float s_dist[WAVES][16 * 17];   // per-wave dist tile, stride 17
  __shared__ float s_sums[WAVES][16][12];    // per-wave voted sums, k = 0..8
  __shared__ int   s_cnt[WAVES][16];         // per-wave radius counts
  __shared__ float s_cd[16][NCAND];          // knn candidate dists
  __shared__ int   s_ci[16][NCAND];          // knn candidate indices

  const int base = blockIdx.x * 16;
  const int tid  = threadIdx.x;
  const int lane = tid & 31;
  const int w    = tid >> 5;
  const int half = lane >> 4;
  const int l16  = lane & 15;

  if (tid < 16) {
    int bi = boundary_indices[base + tid];
    float x = points[3 * bi + 0];
    float y = points[3 * bi + 1];
    float z = points[3 * bi + 2];
    s_bx[tid] = x; s_by[tid] = y; s_bz[tid] = z;
    s_bsq[tid] = x * x + y * y + z * z;
  }
  __syncthreads();

  // A operand of the distance WMMA:  row m = [-2bx, -2by, -2bz, 1]
  v2f a_geo;
  if (half == 0) { a_geo.x = -2.0f * s_bx[l16]; a_geo.y = -2.0f * s_by[l16]; }
  else           { a_geo.x = -2.0f * s_bz[l16]; a_geo.y = 1.0f; }

  // C init = |b|^2 broadcast along the row
  v8f cinit;
#pragma unroll
  for (int j = 0; j < 8; ++j) cinit[j] = s_bsq[j + 8 * half];

  v8f voted;
#pragma unroll
  for (int j = 0; j < 8; ++j) voted[j] = 0.0f;

  int cnt[8];
#pragma unroll
  for (int j = 0; j < 8; ++j) cnt[j] = 0;

  float kd[16]; int ki[16];
#pragma unroll
  for (int s = 0; s < 16; ++s) { kd[s] = 3.0e38f; ki[s] = 0; }

  float* myDist = &s_dist[w][0];

  for (int c0 = w * 16; c0 < NPTS; c0 += CSTRIDE) {
    // B operand (geometry): col n = [px, py, pz, |p|^2]
    int col = c0 + l16;
    float px = points[3 * col + 0];
    float py = points[3 * col + 1];
    float pz = points[3 * col + 2];
    v2f b_geo;
    if (half == 0) { b_geo.x = px; b_geo.y = py; }
    else           { b_geo.x = pz; b_geo.y = px * px + py * py + pz * pz; }

    if (c0 + CSTRIDE < NPTS) __builtin_prefetch(points + 3 * (c0 + CSTRIDE), 0, 0);

    // ---- WMMA #1: 16x16 tile of squared distances ----
    v8f d2 = wmma_f32_16x16x4(a_geo, b_geo, cinit);

    // dist (raw v_sqrt_f32), radius counts, and LDS transpose staging
#pragma unroll
    for (int j = 0; j < 8; ++j) {
      float d = fast_sqrt(fmaxf(d2[j], 1e-12f));
      cnt[j] += (d < RADIUS) ? 1 : 0;
      myDist[(j + 8 * half) * 17 + l16] = d;
    }

    // ---- WMMA #2..#5: voted += W(16x16) @ [part_mask | 1] ----
#pragma unroll
    for (int g = 0; g < 4; ++g) {
      int k0 = g * 4 + half * 2;                 // our two K slots
      float d0 = myDist[l16 * 17 + k0];
      float d1 = myDist[l16 * 17 + k0 + 1];
      v2f a_w;
      a_w.x = (d0 < RADIUS) ? __expf(-d0 * INV_RADIUS) : 0.0f;
      a_w.y = (d1 < RADIUS) ? __expf(-d1 * INV_RADIUS) : 0.0f;
      int p0 = c0 + k0;
      v2f b_pm;
      b_pm.x = (l16 < KPARTS) ? part_mask[p0 * KPARTS + l16]
                              : ((l16 == KPARTS) ? 1.0f : 0.0f);
      b_pm.y = (l16 < KPARTS) ? part_mask[(p0 + 1) * KPARTS + l16]
                              : ((l16 == KPARTS) ? 1.0f : 0.0f);
      voted = wmma_f32_16x16x4(a_w, b_pm, voted);
    }

    // ---- top-16 maintenance: lane = row l16, this half scans 8 columns ----
    float worst = kd[15];
#pragma unroll
    for (int s = 0; s < 8; ++s) {
      int cc = half * 8 + s;
      float d = myDist[l16 * 17 + cc];
      if (d < worst) {
        float cd = d; int cidx = c0 + cc;
#pragma unroll
        for (int t = 0; t < 16; ++t) {
          if (cd < kd[t]) {
            float tf = kd[t]; kd[t] = cd; cd = tf;
            int tn = ki[t]; ki[t] = cidx; cidx = tn;
          }
        }
        worst = kd[15];
      }
    }
  }

  // reduce radius counts across the 16 columns of each half-group
#pragma unroll
  for (int j = 0; j < 8; ++j) {
    int v = cnt[j];
    for (int off = 1; off < 16; off <<= 1) v += __shfl_xor(v, off);
    cnt[j] = v;
  }
  if ((lane & 15) == 0) {
#pragma unroll
    for (int j = 0; j < 8; ++j) s_cnt[w][j + 8 * half] = cnt[j];
  }
  // voted accumulator: lane = part column k, VGPR j = row
  if (l16 <= KPARTS) {
#pragma unroll
    for (int j = 0; j < 8; ++j) s_sums[w][j + 8 * half][l16] = voted[j];
  }
  // dump per-wave top-16 candidates: row l16, slot block w*32 + half*16
#pragma unroll
  for (int s = 0; s < 16; ++s) {
    s_cd[l16][w * 32 + half * 16 + s] = kd[s];
    s_ci[l16][w * 32 + half * 16 + s] = ki[s];
  }
  __syncthreads();

  // ---- final per-row combine: 16 lanes, lane = row ----
  if (tid < 16) {
    int row = tid;
    int m = base + row;

    int total = 0;
#pragma unroll
    for (int ww = 0; ww < WAVES; ++ww) total += s_cnt[ww][row];

    float sums[9];
#pragma unroll
    for (int k = 0; k <= KPARTS; ++k) {
      float acc = 0.0f;
#pragma unroll
      for (int ww = 0; ww < WAVES; ++ww) acc += s_sums[ww][row][k];
      sums[k] = acc;
    }

    // merge NCAND candidates -> global top-16 for this row
    float kd2[16]; int ki2[16];
#pragma unroll
    for (int s = 0; s < 16; ++s) { kd2[s] = 3.0e38f; ki2[s] = 0; }
    for (int s = 0; s < NCAND; ++s) {
      float d = s_cd[row][s];
      if (d < kd2[15]) {
        float cd = d; int cidx = s_ci[row][s];
#pragma unroll
        for (int t = 0; t < 16; ++t) {
          if (cd < kd2[t]) {
            float tf = kd2[t]; kd2[t] = cd; cd = tf;
            int tn = ki2[t]; ki2[t] = cidx; cidx = tn;
          }
        }
      }
    }

    // kNN-branch weighted sums
    float ksum = 0.0f;
    float kacc[KPARTS];
#pragma unroll
    for (int k = 0; k < KPARTS; ++k) kacc[k] = 0.0f;
#pragma unroll
    for (int s = 0; s < 16; ++s) {
      float wgt = __expf(-kd2[s] * INV_RADIUS);
      ksum += wgt;
      const float* pm = part_mask + ki2[s] * KPARTS;
#pragma unroll
      for (int k = 0; k < KPARTS; ++k) kacc[k] += wgt * pm[k];
    }

    bool use_radius = (total >= 16);
    float den = (use_radius ? sums[KPARTS] : ksum) + 1e-8f;
    float v[KPARTS];
#pragma unroll
    for (int k = 0; k < KPARTS; ++k)
      v[k] = (use_radius ? sums[k] : kacc[k]) / den;

    int cl = cluster_labels[m];
    if (cl == -1) {                         // noise points keep raw probs
      int bi = boundary_indices[m];
#pragma unroll
      for (int k = 0; k < KPARTS; ++k) v[k] = part_mask[bi * KPARTS + k];
    }

    // softmax over the 8 parts
    float mx = v[0];
#pragma unroll
    for (int k = 1; k < KPARTS; ++k) mx = fmaxf(mx, v[k]);
    float se = 0.0f;
#pragma unroll
    for (int k = 0; k < KPARTS; ++k) { v[k] = __expf(v[k] - mx); se += v[k]; }
    float inv = 1.0f / se;
#pragma unroll
    for (int k = 0; k < KPARTS; ++k)
      out[NPTS + MB + m * KPARTS + k] = v[k] * inv;

    out[NPTS + m] = (float)boundary_indices[m];   // second output: indices
  }
}

// ---------------------------------------------------------------------------
extern "C" void kernel_launch(void* const* d_in, const int* in_sizes, int n_in,
                              void* d_out, int out_size, void* d_ws, size_t ws_size,
                              hipStream_t stream) {
  const float* points            = (const float*)d_in[0];   // (N,3)
  const float* part_mask         = (const float*)d_in[1];   // (N,8)
  const int*   neighbor_indices  = (const int*)d_in[2];     // (N,16)
  const int*   boundary_indices  = (const int*)d_in[3];     // (M,)
  const int*   cluster_labels    = (const int*)d_in[4];     // (M,)
  float*       out               = (float*)d_out;           // N + M + M*8 floats
  int*         labels            = (int*)d_ws;              // N ints scratch

  labels_kernel<<<(NPTS + 255) / 256, 256, 0, stream>>>(part_mask, labels);
  boundary_kernel<<<(NPTS + 255) / 256, 256, 0, stream>>>(labels, neighbor_indices, out);
  scatter_kernel<<<(MB + 255) / 256, 256, 0, stream>>>(boundary_indices, cluster_labels, out);
  voting_kernel<<<MB / 16, WAVES * 32, 0, stream>>>(points, part_mask,
                                                    boundary_indices, cluster_labels, out);
}